// looc_layer_58643483459942
// MI455X (gfx1250) — compile-verified
//
#include <hip/hip_runtime.h>
#include <hip/hip_bf16.h>
#include <math.h>

// Problem constants (from reference): x (B=1024, D=256), weight (1, D=256, C=1000)
#define BDIM 1024
#define DDIM 256
#define CDIM 1000
#define EPSV 1e-6f

typedef __attribute__((ext_vector_type(2))) float v2f;
typedef __attribute__((ext_vector_type(8))) float v8f;

// ---------------------------------------------------------------------------
// Kernel 1: xterm[b] = sum_d x[b,d]^2 + 2*eps*sum_d x[b,d]
// One wave (32 lanes) per row; each lane handles 8 strided elements (coalesced),
// then a 5-step wave32 xor-shuffle reduction.
// ---------------------------------------------------------------------------
__global__ void pd_xterm_kernel(const float* __restrict__ x,
                                float* __restrict__ xterm) {
    const int lane = threadIdx.x & 31;
    const int wave = threadIdx.x >> 5;
    const int row  = blockIdx.x * 8 + wave;   // 128 blocks * 8 waves = 1024 rows

    const float* xr = x + row * DDIM;
    float s1 = 0.f, s2 = 0.f;
#pragma unroll
    for (int i = 0; i < DDIM / 32; ++i) {
        float v = xr[lane + i * 32];
        s1 += v;
        s2 += v * v;
    }
#pragma unroll
    for (int off = 16; off > 0; off >>= 1) {
        s1 += __shfl_xor(s1, off, 32);
        s2 += __shfl_xor(s2, off, 32);
    }
    if (lane == 0) {
        xterm[row] = s2 + 2.0f * EPSV * s1;
    }
}

// ---------------------------------------------------------------------------
// Kernel 2: wterm[c] = sum_d w[d,c]^2 - 2*eps*sum_d w[d,c] + D*eps^2
// One thread per column; loads w[d*C + c] are fully coalesced across threads.
// ---------------------------------------------------------------------------
__global__ void pd_wterm_kernel(const float* __restrict__ w,
                                float* __restrict__ wterm) {
    const int c = blockIdx.x * blockDim.x + threadIdx.x;
    if (c >= CDIM) return;
    float s1 = 0.f, s2 = 0.f;
#pragma unroll 8
    for (int d = 0; d < DDIM; ++d) {
        float v = w[d * CDIM + c];
        s1 += v;
        s2 += v * v;
    }
    wterm[c] = s2 - 2.0f * EPSV * s1 + (float)DDIM * EPSV * EPSV;
}

// ---------------------------------------------------------------------------
// Kernel 3: WMMA fp32 GEMM  G[b,c] = x_b . w_c  (16x16x4 f32 WMMA, K-chain of 64)
// epilogue:  out = sqrt(max(xterm[b] + wterm[c] - 2*G, 0))
//
// Block = 256 threads = 8 wave32 waves arranged 2 (M) x 4 (N), each wave owns a
// 16x16 tile -> block macro-tile 32 (M) x 64 (N). Grid = (16, 32) covers
// 1024 x 1024; columns >= 1000 are address-clamped on load and masked on store.
// ---------------------------------------------------------------------------
__global__ void pd_wmma_kernel(const float* __restrict__ x,
                               const float* __restrict__ w,
                               const float* __restrict__ xterm,
                               const float* __restrict__ wterm,
                               float* __restrict__ out) {
    const int lane  = threadIdx.x & 31;
    const int wave  = threadIdx.x >> 5;
    const int wm    = wave >> 2;          // 0..1
    const int wn    = wave & 3;           // 0..3

    const int tile_m0 = blockIdx.y * 32 + wm * 16;
    const int tile_n0 = blockIdx.x * 64 + wn * 16;

    const int l15   = lane & 15;
    const int khalf = (lane >> 4) * 2;    // 0 or 2: which K-pair this half-wave holds

    // A operand: row of x this lane feeds
    const float* arow = x + (tile_m0 + l15) * DDIM + khalf;
    // B operand: column of w this lane feeds (clamped; garbage cols never stored)
    const int col   = tile_n0 + l15;
    const int colc  = col < CDIM ? col : (CDIM - 1);
    const float* bcol = w + colc;

    v8f acc = {};

#pragma unroll 8
    for (int k = 0; k < DDIM; k += 4) {
        v2f a, b;
        // contiguous pair -> single global_load_b64
        a.x = arow[k];
        a.y = arow[k + 1];
        // strided by C across K, coalesced across lanes
        b.x = bcol[(k + khalf) * CDIM];
        b.y = bcol[(k + khalf + 1) * CDIM];
        acc = __builtin_amdgcn_wmma_f32_16x16x4_f32(
            /*neg_a=*/false, a, /*neg_b=*/false, b,
            /*c_mod=*/(short)0, acc, /*reuse_a=*/false, /*reuse_b=*/false);
    }

    // Epilogue: C/D layout -> (M = r + 8*(lane>>4), N = lane&15)
    const float wt  = wterm[colc];
    const int   mhi = (lane >> 4) * 8;
#pragma unroll
    for (int r = 0; r < 8; ++r) {
        const int row_g = tile_m0 + r + mhi;
        const float s   = xterm[row_g] + wt - 2.0f * acc[r];
        if (col < CDIM) {
            out[row_g * CDIM + col] = sqrtf(fmaxf(s, 0.0f));
        }
    }
}

// ---------------------------------------------------------------------------
extern "C" void kernel_launch(void* const* d_in, const int* in_sizes, int n_in,
                              void* d_out, int out_size, void* d_ws, size_t ws_size,
                              hipStream_t stream) {
    const float* x = (const float*)d_in[0];   // (1024, 256)
    const float* w = (const float*)d_in[1];   // (1, 256, 1000)
    float* out = (float*)d_out;               // (1024, 1000)

    float* xterm = (float*)d_ws;              // 1024 floats
    float* wterm = xterm + BDIM;              // 1000 floats

    // xterm: 1024 rows, 8 waves (256 thr) per block -> 128 blocks
    pd_xterm_kernel<<<BDIM / 8, 256, 0, stream>>>(x, xterm);
    // wterm: 1000 columns, 256 thr per block -> 4 blocks
    pd_wterm_kernel<<<(CDIM + 255) / 256, 256, 0, stream>>>(w, wterm);
    // GEMM + epilogue: grid (16, 32) of 32x64 macro-tiles
    dim3 grid(16, 32, 1);
    pd_wmma_kernel<<<grid, 256, 0, stream>>>(x, w, xterm, wterm, out);
}